// PhaseMLP_10445360463873
// MI455X (gfx1250) — compile-verified
//
#include <hip/hip_runtime.h>
#include <math.h>

typedef __attribute__((ext_vector_type(2))) float v2f;
typedef __attribute__((ext_vector_type(4))) float v4f;
typedef __attribute__((ext_vector_type(8))) float v8f;

// ---------------------------------------------------------------------------
// Per-sample Catmull-Rom coefficients routed onto the 4 phase anchors.
// weight(b) = sum_p a_p(b) * W_p  (i0..i3 are always a permutation of 0..3).
// Same coefficients apply to the bias.
// ---------------------------------------------------------------------------
__global__ void phase_coef_kernel(const float* __restrict__ phase,
                                  float* __restrict__ coef, int n)
{
    int b = blockIdx.x * blockDim.x + threadIdx.x;
    if (b >= n) return;
    float t  = 4.0f * phase[b];
    float ft = floorf(t);
    int   i1 = ((int)ft) & 3;
    float w  = t - ft;
    float w2 = w * w;
    float w3 = w2 * w;
    float c_im1 = -0.5f * w + w2 - 0.5f * w3;        // -> anchor i1-1
    float c_i   = 1.0f - 2.5f * w2 + 1.5f * w3;      // -> anchor i1
    float c_ip1 = 0.5f * w + 2.0f * w2 - 1.5f * w3;  // -> anchor i1+1
    float c_ip2 = -0.5f * w2 + 0.5f * w3;            // -> anchor i1+2
    float byoff[4] = {c_i, c_ip1, c_ip2, c_im1};     // indexed by (p - i1) mod 4
#pragma unroll
    for (int p = 0; p < 4; ++p)
        coef[b * 4 + p] = byoff[(p - i1) & 3];
}

// ---------------------------------------------------------------------------
// W [4][I][O] -> Wt [4][O][I]  so a lane's B-fragment (fixed N, consecutive K)
// is one contiguous b64 load.
// ---------------------------------------------------------------------------
__global__ void transpose_w_kernel(const float* __restrict__ W,
                                   float* __restrict__ Wt, int I, int O)
{
    int idx = blockIdx.x * blockDim.x + threadIdx.x;
    int total = 4 * I * O;
    if (idx >= total) return;
    int p = idx / (I * O);
    int r = idx - p * (I * O);
    int i = r / O;
    int o = r - i * O;
    Wt[((size_t)p * O + o) * I + i] = W[idx];
}

// ---------------------------------------------------------------------------
// Fused layer: Y[b,o] = act( sum_p a_p(b) * ((X @ W_p)[b,o] + bias_p[o]) )
// Each wave owns a 64(M) x 16(N) output slab: 4 M-tiles x 4 anchor
// accumulators = 16 f32 WMMA per 4-k chunk, fed by 8 b64 loads.
//
// A 16x4 f32 frag: lane L holds X[M=L&15][K=2*(L>>4)+j], j=0,1
// B 4x16  f32 frag: lane L holds W[K=2*(L>>4)+j][N=L&15]   (from Wt, contiguous)
// C/D 16x16 f32:    VGPR v, lane L -> row = v + 8*(L>>4), col = L&15
// ---------------------------------------------------------------------------
template <int I, int O, bool ELU>
__global__ __launch_bounds__(128) void phase_mlp_layer(
    const float* __restrict__ X,     // [B][I]
    const float* __restrict__ Wt,    // [4][O][I] (transposed)
    const float* __restrict__ Bias,  // [4][1][O]
    const float* __restrict__ coef,  // [B][4]
    float* __restrict__ Y)           // [B][O]
{
    const int lane  = threadIdx.x & 31;
    const int wave  = threadIdx.x >> 5;
    const int mbase = blockIdx.x << 6;                 // 64 rows per block
    const int nbase = (blockIdx.y * 4 + wave) << 4;    // 16 cols per wave

    const int rc = lane & 15;  // A-row / B-col / C-col within tile
    const int kh = lane >> 4;  // K-half of fragment

    v8f acc[4][4];             // [anchor][mtile]
#pragma unroll
    for (int p = 0; p < 4; ++p)
#pragma unroll
        for (int m = 0; m < 4; ++m)
            acc[p][m] = (v8f){};

    const float* xa = X + (size_t)(mbase + rc) * I + (kh << 1);
    const float* wt0 = Wt + ((size_t)0 * O + nbase + rc) * I + (kh << 1);
    const float* wt1 = Wt + ((size_t)1 * O + nbase + rc) * I + (kh << 1);
    const float* wt2 = Wt + ((size_t)2 * O + nbase + rc) * I + (kh << 1);
    const float* wt3 = Wt + ((size_t)3 * O + nbase + rc) * I + (kh << 1);

#pragma unroll 2
    for (int k = 0; k < I; k += 4) {
        v2f a[4];
#pragma unroll
        for (int m = 0; m < 4; ++m)
            a[m] = *(const v2f*)(xa + (size_t)m * 16 * I + k);
        v2f b[4];
        b[0] = *(const v2f*)(wt0 + k);
        b[1] = *(const v2f*)(wt1 + k);
        b[2] = *(const v2f*)(wt2 + k);
        b[3] = *(const v2f*)(wt3 + k);
#pragma unroll
        for (int p = 0; p < 4; ++p)
#pragma unroll
            for (int m = 0; m < 4; ++m)
                acc[p][m] = __builtin_amdgcn_wmma_f32_16x16x4_f32(
                    false, a[m], false, b[p], (short)0, acc[p][m], false, false);
    }

    float bias[4];
#pragma unroll
    for (int p = 0; p < 4; ++p)
        bias[p] = Bias[p * O + nbase + rc];

#pragma unroll
    for (int m = 0; m < 4; ++m) {
#pragma unroll
        for (int v = 0; v < 8; ++v) {
            int row = mbase + (m << 4) + v + (kh << 3);
            v4f cf = *(const v4f*)(coef + (size_t)row * 4);
            float val = cf[0] * (acc[0][m][v] + bias[0])
                      + cf[1] * (acc[1][m][v] + bias[1])
                      + cf[2] * (acc[2][m][v] + bias[2])
                      + cf[3] * (acc[3][m][v] + bias[3]);
            if (ELU) val = (val > 0.0f) ? val : (expf(val) - 1.0f);
            Y[(size_t)row * O + nbase + rc] = val;
        }
    }
}

extern "C" void kernel_launch(void* const* d_in, const int* in_sizes, int n_in,
                              void* d_out, int out_size, void* d_ws, size_t ws_size,
                              hipStream_t stream)
{
    const float* x     = (const float*)d_in[0];
    const float* phase = (const float*)d_in[1];
    const float* w0    = (const float*)d_in[2];
    const float* b0    = (const float*)d_in[3];
    const float* w1    = (const float*)d_in[4];
    const float* b1    = (const float*)d_in[5];
    const float* w2    = (const float*)d_in[6];
    const float* b2    = (const float*)d_in[7];
    float*       out   = (float*)d_out;

    const int B = in_sizes[1];  // 4096 (one phase scalar per sample)
    constexpr int D0 = 128, D1 = 256, D2 = 256, D3 = 128;

    char*  ws   = (char*)d_ws;
    float* coef = (float*)ws;                                    // B*4
    float* act0 = coef + (size_t)B * 4;                          // B*D1
    float* act1 = act0 + (size_t)B * D1;                         // B*D2
    float* wt0  = act1 + (size_t)B * D2;                         // 4*D1*D0
    float* wt1  = wt0 + (size_t)4 * D0 * D1;                     // 4*D2*D1
    float* wt2  = wt1 + (size_t)4 * D1 * D2;                     // 4*D3*D2

    phase_coef_kernel<<<(B + 255) / 256, 256, 0, stream>>>(phase, coef, B);

    transpose_w_kernel<<<(4 * D0 * D1 + 255) / 256, 256, 0, stream>>>(w0, wt0, D0, D1);
    transpose_w_kernel<<<(4 * D1 * D2 + 255) / 256, 256, 0, stream>>>(w1, wt1, D1, D2);
    transpose_w_kernel<<<(4 * D2 * D3 + 255) / 256, 256, 0, stream>>>(w2, wt2, D2, D3);

    dim3 blk(128);
    phase_mlp_layer<D0, D1, true ><<<dim3(B / 64, D1 / 64), blk, 0, stream>>>(x,    wt0, b0, coef, act0);
    phase_mlp_layer<D1, D2, true ><<<dim3(B / 64, D2 / 64), blk, 0, stream>>>(act0, wt1, b1, coef, act1);
    phase_mlp_layer<D2, D3, false><<<dim3(B / 64, D3 / 64), blk, 0, stream>>>(act1, wt2, b2, coef, out);
}